// HeteroConvBlock_5428838662528
// MI455X (gfx1250) — compile-verified
//
#include <hip/hip_runtime.h>
#include <math.h>

#define HID   128
#define NH    4
#define DH    32
#define NBALL 60000
#define NPLY  2000
#define NCTX  4000
#define EM_N  150000
#define EP_N  300000
#define EI_N  150000

typedef __attribute__((ext_vector_type(2))) float v2f;
typedef __attribute__((ext_vector_type(8))) float v8f;

static __device__ __forceinline__ void atomicMaxF(float* addr, float val) {
    int* ia = reinterpret_cast<int*>(addr);
    int cur = *ia;
    while (__int_as_float(cur) < val) {
        int prev = atomicCAS(ia, cur, __float_as_int(val));
        if (prev == cur) break;
        cur = prev;
    }
}

// ---------------------------------------------------------------------------
// Generic fill
// ---------------------------------------------------------------------------
__global__ void fill_f32(float* __restrict__ p, float v, size_t n) {
    size_t i = (size_t)blockIdx.x * blockDim.x + threadIdx.x;
    size_t stride = (size_t)gridDim.x * blockDim.x;
    for (; i < n; i += stride) p[i] = v;
}

// ---------------------------------------------------------------------------
// Generic C[M,128] = A[M,128] @ B[128,128] (+ bias) (+ add0) (+ add1)
// One wave per 16x16 C tile, V_WMMA_F32_16X16X4_F32 (fp32 exact).
// ---------------------------------------------------------------------------
__global__ __launch_bounds__(256) void gemm128_wmma(
    const float* __restrict__ A, const float* __restrict__ B,
    const float* __restrict__ bias,
    const float* __restrict__ add0, const float* __restrict__ add1,
    float* __restrict__ C, int M)
{
    const int wave = threadIdx.x >> 5;
    const int lane = threadIdx.x & 31;
    const int m0 = blockIdx.x * 16;
    const int n0 = wave * 16;
    if (m0 >= M) return;

    const int r  = lane & 15;
    const int kh = lane >> 4;

    const float* arow = A + (size_t)(m0 + r) * HID;
    v8f c = {0.f, 0.f, 0.f, 0.f, 0.f, 0.f, 0.f, 0.f};

    for (int kb = 0; kb < HID; kb += 4) {
        v2f a, b;
        a.x = arow[kb + kh * 2 + 0];
        a.y = arow[kb + kh * 2 + 1];
        const size_t off = (size_t)(kb + kh * 2) * HID + n0 + r;
        b.x = B[off];
        b.y = B[off + HID];
        c = __builtin_amdgcn_wmma_f32_16x16x4_f32(
                false, a, false, b, (short)0, c, false, false);
    }

    const int col = n0 + r;
    const float bv = bias ? bias[col] : 0.0f;
#pragma unroll
    for (int i = 0; i < 8; ++i) {
        const int row = m0 + i + kh * 8;
        size_t idx = (size_t)row * HID + col;
        float val = c[i] + bv;
        if (add0) val += add0[idx];
        if (add1) val += add1[idx];
        C[idx] = val;
    }
}

// ---------------------------------------------------------------------------
// Fused 5-output GEMM for the ball node features: each A fragment is loaded
// once and feeds 5 WMMA accumulator chains (xr, q, k, v, skip).
// Cuts redundant x_ball reads 5x vs separate GEMMs.
// ---------------------------------------------------------------------------
__global__ __launch_bounds__(256) void gemm128_wmma_ball(
    const float* __restrict__ A,
    const float* __restrict__ Bwr, const float* __restrict__ Bq,
    const float* __restrict__ Bk,  const float* __restrict__ Bv,
    const float* __restrict__ Bs,
    const float* __restrict__ biasq, const float* __restrict__ biask,
    const float* __restrict__ biasv, const float* __restrict__ biass,
    float* __restrict__ Cwr, float* __restrict__ Cq,
    float* __restrict__ Ck,  float* __restrict__ Cv,
    float* __restrict__ Cs,  int M)
{
    const int wave = threadIdx.x >> 5;
    const int lane = threadIdx.x & 31;
    const int m0 = blockIdx.x * 16;
    const int n0 = wave * 16;
    if (m0 >= M) return;

    const int r  = lane & 15;
    const int kh = lane >> 4;

    const float* arow = A + (size_t)(m0 + r) * HID;
    v8f c0 = {0.f,0.f,0.f,0.f,0.f,0.f,0.f,0.f};
    v8f c1 = c0, c2 = c0, c3 = c0, c4 = c0;

    for (int kb = 0; kb < HID; kb += 4) {
        v2f a;
        a.x = arow[kb + kh * 2 + 0];
        a.y = arow[kb + kh * 2 + 1];
        const size_t off = (size_t)(kb + kh * 2) * HID + n0 + r;
        v2f b;
        b.x = Bwr[off]; b.y = Bwr[off + HID];
        c0 = __builtin_amdgcn_wmma_f32_16x16x4_f32(false, a, false, b, (short)0, c0, false, false);
        b.x = Bq[off];  b.y = Bq[off + HID];
        c1 = __builtin_amdgcn_wmma_f32_16x16x4_f32(false, a, false, b, (short)0, c1, false, false);
        b.x = Bk[off];  b.y = Bk[off + HID];
        c2 = __builtin_amdgcn_wmma_f32_16x16x4_f32(false, a, false, b, (short)0, c2, false, false);
        b.x = Bv[off];  b.y = Bv[off + HID];
        c3 = __builtin_amdgcn_wmma_f32_16x16x4_f32(false, a, false, b, (short)0, c3, false, false);
        b.x = Bs[off];  b.y = Bs[off + HID];
        c4 = __builtin_amdgcn_wmma_f32_16x16x4_f32(false, a, false, b, (short)0, c4, false, false);
    }

    const int col = n0 + r;
    const float vq = biasq[col], vk = biask[col], vv = biasv[col], vs = biass[col];
#pragma unroll
    for (int i = 0; i < 8; ++i) {
        const int row = m0 + i + kh * 8;
        size_t idx = (size_t)row * HID + col;
        Cwr[idx] = c0[i];
        Cq[idx]  = c1[i] + vq;
        Ck[idx]  = c2[i] + vk;
        Cv[idx]  = c3[i] + vv;
        Cs[idx]  = c4[i] + vs;
    }
}

// ---------------------------------------------------------------------------
// GATv2 edge pass 1: logit = sum_d leaky(xl[src]+xr[dst]) * att ; atomic max
// ---------------------------------------------------------------------------
__global__ void gat_pass1(const int* __restrict__ em,
                          const float* __restrict__ xl,
                          const float* __restrict__ xr,
                          const float* __restrict__ att,
                          float* __restrict__ m_g)
{
    int idx = blockIdx.x * blockDim.x + threadIdx.x;
    if (idx >= EM_N * NH) return;
    int e = idx >> 2, h = idx & 3;
    int src = em[e], dst = em[EM_N + e];
    const float4* pl = (const float4*)(xl + (size_t)src * HID + h * DH);
    const float4* pr = (const float4*)(xr + (size_t)dst * HID + h * DH);
    const float4* pa = (const float4*)(att + h * DH);
    float logit = 0.f;
#pragma unroll
    for (int i = 0; i < 8; ++i) {
        float4 a = pl[i], b = pr[i], w = pa[i];
        float t;
        t = a.x + b.x; t = t > 0.f ? t : 0.2f * t; logit += t * w.x;
        t = a.y + b.y; t = t > 0.f ? t : 0.2f * t; logit += t * w.y;
        t = a.z + b.z; t = t > 0.f ? t : 0.2f * t; logit += t * w.z;
        t = a.w + b.w; t = t > 0.f ? t : 0.2f * t; logit += t * w.w;
    }
    atomicMaxF(&m_g[dst * NH + h], logit);
}

__global__ void gat_pass2(const int* __restrict__ em,
                          const float* __restrict__ xl,
                          const float* __restrict__ xr,
                          const float* __restrict__ att,
                          const float* __restrict__ m_g,
                          float* __restrict__ s_g,
                          float* __restrict__ num_g)
{
    int idx = blockIdx.x * blockDim.x + threadIdx.x;
    if (idx >= EM_N * NH) return;
    int e = idx >> 2, h = idx & 3;
    int src = em[e], dst = em[EM_N + e];
    const float4* pl = (const float4*)(xl + (size_t)src * HID + h * DH);
    const float4* pr = (const float4*)(xr + (size_t)dst * HID + h * DH);
    const float4* pa = (const float4*)(att + h * DH);
    float4 av[8];
    float logit = 0.f;
#pragma unroll
    for (int i = 0; i < 8; ++i) {
        float4 a = pl[i], b = pr[i], w = pa[i];
        av[i] = a;
        float t;
        t = a.x + b.x; t = t > 0.f ? t : 0.2f * t; logit += t * w.x;
        t = a.y + b.y; t = t > 0.f ? t : 0.2f * t; logit += t * w.y;
        t = a.z + b.z; t = t > 0.f ? t : 0.2f * t; logit += t * w.z;
        t = a.w + b.w; t = t > 0.f ? t : 0.2f * t; logit += t * w.w;
    }
    float wgt = expf(logit - m_g[dst * NH + h]);
    atomicAdd(&s_g[dst * NH + h], wgt);
    float* po = num_g + (size_t)dst * HID + h * DH;
#pragma unroll
    for (int i = 0; i < 8; ++i) {
        atomicAdd(po + 4 * i + 0, wgt * av[i].x);
        atomicAdd(po + 4 * i + 1, wgt * av[i].y);
        atomicAdd(po + 4 * i + 2, wgt * av[i].z);
        atomicAdd(po + 4 * i + 3, wgt * av[i].w);
    }
}

// ---------------------------------------------------------------------------
// TransformerConv edge passes. ee = ea_p[e] * t_We (edge_dim=1).
// ---------------------------------------------------------------------------
__global__ void tr_pass1(const int* __restrict__ ep,
                         const float* __restrict__ ea_p,
                         const float* __restrict__ q,
                         const float* __restrict__ k,
                         const float* __restrict__ we,
                         float* __restrict__ m_t)
{
    int idx = blockIdx.x * blockDim.x + threadIdx.x;
    if (idx >= EP_N * NH) return;
    int e = idx >> 2, h = idx & 3;
    int src = ep[e], dst = ep[EP_N + e];
    float ea = ea_p[e];
    const float4* pq = (const float4*)(q + (size_t)dst * HID + h * DH);
    const float4* pk = (const float4*)(k + (size_t)src * HID + h * DH);
    const float4* pw = (const float4*)(we + h * DH);
    float logit = 0.f;
#pragma unroll
    for (int i = 0; i < 8; ++i) {
        float4 a = pq[i], b = pk[i], w = pw[i];
        logit += a.x * (b.x + ea * w.x);
        logit += a.y * (b.y + ea * w.y);
        logit += a.z * (b.z + ea * w.z);
        logit += a.w * (b.w + ea * w.w);
    }
    logit *= 0.17677669529663688f; // 1/sqrt(32)
    atomicMaxF(&m_t[dst * NH + h], logit);
}

__global__ void tr_pass2(const int* __restrict__ ep,
                         const float* __restrict__ ea_p,
                         const float* __restrict__ q,
                         const float* __restrict__ k,
                         const float* __restrict__ v,
                         const float* __restrict__ we,
                         const float* __restrict__ m_t,
                         float* __restrict__ s_t,
                         float* __restrict__ num_t)
{
    int idx = blockIdx.x * blockDim.x + threadIdx.x;
    if (idx >= EP_N * NH) return;
    int e = idx >> 2, h = idx & 3;
    int src = ep[e], dst = ep[EP_N + e];
    float ea = ea_p[e];
    const float4* pq = (const float4*)(q + (size_t)dst * HID + h * DH);
    const float4* pk = (const float4*)(k + (size_t)src * HID + h * DH);
    const float4* pv = (const float4*)(v + (size_t)src * HID + h * DH);
    const float4* pw = (const float4*)(we + h * DH);
    float logit = 0.f;
    float4 wv[8];
#pragma unroll
    for (int i = 0; i < 8; ++i) {
        float4 a = pq[i], b = pk[i], w = pw[i];
        wv[i] = w;
        logit += a.x * (b.x + ea * w.x);
        logit += a.y * (b.y + ea * w.y);
        logit += a.z * (b.z + ea * w.z);
        logit += a.w * (b.w + ea * w.w);
    }
    logit *= 0.17677669529663688f;
    float wgt = expf(logit - m_t[dst * NH + h]);
    atomicAdd(&s_t[dst * NH + h], wgt);
    float* po = num_t + (size_t)dst * HID + h * DH;
#pragma unroll
    for (int i = 0; i < 8; ++i) {
        float4 val = pv[i];
        atomicAdd(po + 4 * i + 0, wgt * (val.x + ea * wv[i].x));
        atomicAdd(po + 4 * i + 1, wgt * (val.y + ea * wv[i].y));
        atomicAdd(po + 4 * i + 2, wgt * (val.z + ea * wv[i].z));
        atomicAdd(po + 4 * i + 3, wgt * (val.w + ea * wv[i].w));
    }
}

// ---------------------------------------------------------------------------
// SAGE scatter: sum x_ball[src] into ctx_sum[dst]; count per dst.
// ---------------------------------------------------------------------------
__global__ void sage_count(const int* __restrict__ ei, float* __restrict__ cnt) {
    int e = blockIdx.x * blockDim.x + threadIdx.x;
    if (e >= EI_N) return;
    atomicAdd(&cnt[ei[EI_N + e]], 1.0f);
}

__global__ void sage_scatter(const int* __restrict__ ei,
                             const float* __restrict__ xball,
                             float* __restrict__ ctx_sum)
{
    int idx = blockIdx.x * blockDim.x + threadIdx.x;
    if (idx >= EI_N * NH) return;
    int e = idx >> 2, c = idx & 3;
    int src = ei[e], dst = ei[EI_N + e];
    const float4* px = (const float4*)(xball + (size_t)src * HID + c * DH);
    float* po = ctx_sum + (size_t)dst * HID + c * DH;
#pragma unroll
    for (int i = 0; i < 8; ++i) {
        float4 val = px[i];
        atomicAdd(po + 4 * i + 0, val.x);
        atomicAdd(po + 4 * i + 1, val.y);
        atomicAdd(po + 4 * i + 2, val.z);
        atomicAdd(po + 4 * i + 3, val.w);
    }
}

__global__ void sage_mean(const float* __restrict__ sum,
                          const float* __restrict__ cnt,
                          float* __restrict__ mean)
{
    int i = blockIdx.x * blockDim.x + threadIdx.x;
    if (i >= NCTX * HID) return;
    mean[i] = sum[i] / fmaxf(cnt[i >> 7], 1.0f);
}

// ---------------------------------------------------------------------------
// Ball finalize: pre = num_g/s_g + g_b + num_t/s_t + skip + x_ball; LayerNorm.
// One wave32 per row; each lane owns 4 contiguous columns (one head chunk).
// ---------------------------------------------------------------------------
__global__ __launch_bounds__(32) void ball_finalize(
    const float* __restrict__ num_g, const float* __restrict__ s_g,
    const float* __restrict__ num_t, const float* __restrict__ s_t,
    const float* __restrict__ skipb, const float* __restrict__ xball,
    const float* __restrict__ g_b,
    const float* __restrict__ ln_g, const float* __restrict__ ln_b,
    float* __restrict__ out)
{
    int row  = blockIdx.x;
    int lane = threadIdx.x;
    int h    = lane >> 3;
    size_t base = (size_t)row * HID + lane * 4;
    float sg = s_g[row * NH + h] + 1e-16f;
    float st = s_t[row * NH + h] + 1e-16f;
    float4 ng = *(const float4*)(num_g + base);
    float4 nt = *(const float4*)(num_t + base);
    float4 sk = *(const float4*)(skipb + base);
    float4 xb = *(const float4*)(xball + base);
    float4 gb = *(const float4*)(g_b + lane * 4);
    float pre[4];
    pre[0] = ng.x / sg + gb.x + nt.x / st + sk.x + xb.x;
    pre[1] = ng.y / sg + gb.y + nt.y / st + sk.y + xb.y;
    pre[2] = ng.z / sg + gb.z + nt.z / st + sk.z + xb.z;
    pre[3] = ng.w / sg + gb.w + nt.w / st + sk.w + xb.w;
    float sum = pre[0] + pre[1] + pre[2] + pre[3];
#pragma unroll
    for (int off = 16; off > 0; off >>= 1) sum += __shfl_xor(sum, off, 32);
    float mu = sum * (1.0f / HID);
    float var = 0.f;
#pragma unroll
    for (int i = 0; i < 4; ++i) { float d = pre[i] - mu; var += d * d; }
#pragma unroll
    for (int off = 16; off > 0; off >>= 1) var += __shfl_xor(var, off, 32);
    var *= (1.0f / HID);
    float inv = rsqrtf(var + 1e-5f);
    float4 lg = *(const float4*)(ln_g + lane * 4);
    float4 lb = *(const float4*)(ln_b + lane * 4);
    float4 o;
    o.x = (pre[0] - mu) * inv * lg.x + lb.x;
    o.y = (pre[1] - mu) * inv * lg.y + lb.y;
    o.z = (pre[2] - mu) * inv * lg.z + lb.z;
    o.w = (pre[3] - mu) * inv * lg.w + lb.w;
    *(float4*)(out + base) = o;
}

// Plain LayerNorm over precomputed rows (context path).
__global__ __launch_bounds__(32) void ln_rows(
    const float* __restrict__ in,
    const float* __restrict__ g, const float* __restrict__ b,
    float* __restrict__ out)
{
    int row  = blockIdx.x;
    int lane = threadIdx.x;
    size_t base = (size_t)row * HID + lane * 4;
    float4 xv = *(const float4*)(in + base);
    float pre[4] = {xv.x, xv.y, xv.z, xv.w};
    float sum = pre[0] + pre[1] + pre[2] + pre[3];
#pragma unroll
    for (int off = 16; off > 0; off >>= 1) sum += __shfl_xor(sum, off, 32);
    float mu = sum * (1.0f / HID);
    float var = 0.f;
#pragma unroll
    for (int i = 0; i < 4; ++i) { float d = pre[i] - mu; var += d * d; }
#pragma unroll
    for (int off = 16; off > 0; off >>= 1) var += __shfl_xor(var, off, 32);
    var *= (1.0f / HID);
    float inv = rsqrtf(var + 1e-5f);
    float4 lg = *(const float4*)(g + lane * 4);
    float4 lb = *(const float4*)(b + lane * 4);
    float4 o;
    o.x = (pre[0] - mu) * inv * lg.x + lb.x;
    o.y = (pre[1] - mu) * inv * lg.y + lb.y;
    o.z = (pre[2] - mu) * inv * lg.z + lb.z;
    o.w = (pre[3] - mu) * inv * lg.w + lb.w;
    *(float4*)(out + base) = o;
}

__global__ void copy_f4(const float4* __restrict__ in, float4* __restrict__ out, int n) {
    int i = blockIdx.x * blockDim.x + threadIdx.x;
    if (i < n) out[i] = in[i];
}

// ---------------------------------------------------------------------------
extern "C" void kernel_launch(void* const* d_in, const int* in_sizes, int n_in,
                              void* d_out, int out_size, void* d_ws, size_t ws_size,
                              hipStream_t stream) {
    const float* x_ball    = (const float*)d_in[0];
    const float* x_player  = (const float*)d_in[1];
    const float* x_context = (const float*)d_in[2];
    const int*   em        = (const int*)d_in[3];
    const int*   ep        = (const int*)d_in[4];
    const int*   ei        = (const int*)d_in[5];
    const float* ea_p      = (const float*)d_in[6];
    const float* g_Wl      = (const float*)d_in[7];
    const float* g_Wr      = (const float*)d_in[8];
    const float* g_att     = (const float*)d_in[9];
    const float* g_b       = (const float*)d_in[10];
    const float* t_Wq      = (const float*)d_in[11];
    const float* t_bq      = (const float*)d_in[12];
    const float* t_Wk      = (const float*)d_in[13];
    const float* t_bk      = (const float*)d_in[14];
    const float* t_Wv      = (const float*)d_in[15];
    const float* t_bv      = (const float*)d_in[16];
    const float* t_We      = (const float*)d_in[17];
    const float* t_Wskip   = (const float*)d_in[18];
    const float* t_bskip   = (const float*)d_in[19];
    const float* s_Wl      = (const float*)d_in[20];
    const float* s_bl      = (const float*)d_in[21];
    const float* s_Wr      = (const float*)d_in[22];
    const float* ln_ball_g = (const float*)d_in[23];
    const float* ln_ball_b = (const float*)d_in[24];
    const float* ln_ctx_g  = (const float*)d_in[25];
    const float* ln_ctx_b  = (const float*)d_in[26];
    float* out = (float*)d_out;

    // Workspace layout (floats)
    float* W = (float*)d_ws;
    size_t o = 0;
    float* xl      = W + o; o += (size_t)NPLY * HID;
    float* xr      = W + o; o += (size_t)NBALL * HID;
    float* qb      = W + o; o += (size_t)NBALL * HID;
    float* kb      = W + o; o += (size_t)NBALL * HID;
    float* vb      = W + o; o += (size_t)NBALL * HID;
    float* skipb   = W + o; o += (size_t)NBALL * HID;
    float* num_g   = W + o; o += (size_t)NBALL * HID;   // zero region start
    float* num_t   = W + o; o += (size_t)NBALL * HID;
    float* s_g     = W + o; o += (size_t)NBALL * NH;
    float* s_t     = W + o; o += (size_t)NBALL * NH;
    float* ctx_sum = W + o; o += (size_t)NCTX * HID;
    float* ctx_cnt = W + o; o += (size_t)NCTX;          // zero region end
    float* ctx_buf = W + o; o += (size_t)NCTX * HID;
    float* meanb   = W + o; o += (size_t)NCTX * HID;
    float* m_g     = W + o; o += (size_t)NBALL * NH;    // -inf region
    float* m_t     = W + o; o += (size_t)NBALL * NH;

    const size_t zeroN = 2 * (size_t)NBALL * HID + 2 * (size_t)NBALL * NH
                       + (size_t)NCTX * HID + NCTX;
    const size_t infN  = 2 * (size_t)NBALL * NH;

    fill_f32<<<2048, 256, 0, stream>>>(num_g, 0.0f, zeroN);
    fill_f32<<<512, 256, 0, stream>>>(m_g, -INFINITY, infN);

    // Dense transforms (WMMA fp32)
    gemm128_wmma<<<NPLY / 16, 256, 0, stream>>>(x_player, g_Wl, nullptr, nullptr, nullptr, xl, NPLY);
    // Fused: xr, q, k, v, skip share A = x_ball (A loaded once, 5 WMMA chains)
    gemm128_wmma_ball<<<NBALL / 16, 256, 0, stream>>>(
        x_ball, g_Wr, t_Wq, t_Wk, t_Wv, t_Wskip,
        t_bq, t_bk, t_bv, t_bskip,
        xr, qb, kb, vb, skipb, NBALL);

    // GATv2 edges
    {
        int n = EM_N * NH, blk = 256, grd = (n + blk - 1) / blk;
        gat_pass1<<<grd, blk, 0, stream>>>(em, xl, xr, g_att, m_g);
        gat_pass2<<<grd, blk, 0, stream>>>(em, xl, xr, g_att, m_g, s_g, num_g);
    }
    // Transformer edges
    {
        int n = EP_N * NH, blk = 256, grd = (n + blk - 1) / blk;
        tr_pass1<<<grd, blk, 0, stream>>>(ep, ea_p, qb, kb, t_We, m_t);
        tr_pass2<<<grd, blk, 0, stream>>>(ep, ea_p, qb, kb, vb, t_We, m_t, s_t, num_t);
    }
    // SAGE
    {
        int blk = 256;
        sage_count<<<(EI_N + blk - 1) / blk, blk, 0, stream>>>(ei, ctx_cnt);
        int n = EI_N * NH;
        sage_scatter<<<(n + blk - 1) / blk, blk, 0, stream>>>(ei, x_ball, ctx_sum);
        int nm = NCTX * HID;
        sage_mean<<<(nm + blk - 1) / blk, blk, 0, stream>>>(ctx_sum, ctx_cnt, meanb);
    }
    // ctx = LN(mean@s_Wl + s_bl + x_context@s_Wr + x_context)
    gemm128_wmma<<<NCTX / 16, 256, 0, stream>>>(meanb, s_Wl, s_bl, nullptr, nullptr, ctx_buf, NCTX);
    gemm128_wmma<<<NCTX / 16, 256, 0, stream>>>(x_context, s_Wr, nullptr, ctx_buf, x_context, ctx_buf, NCTX);

    // Finalize
    ball_finalize<<<NBALL, 32, 0, stream>>>(num_g, s_g, num_t, s_t, skipb, x_ball,
                                            g_b, ln_ball_g, ln_ball_b, out);
    ln_rows<<<NCTX, 32, 0, stream>>>(ctx_buf, ln_ctx_g, ln_ctx_b,
                                     out + (size_t)NBALL * HID);
    {
        int n = (NPLY * HID) / 4, blk = 256;
        copy_f4<<<(n + blk - 1) / blk, blk, 0, stream>>>(
            (const float4*)x_player,
            (float4*)(out + (size_t)(NBALL + NCTX) * HID), n);
    }
    (void)in_sizes; (void)n_in; (void)out_size; (void)ws_size;
}